// GaussianPointMatcher_79259326480597
// MI455X (gfx1250) — compile-verified
//
#include <hip/hip_runtime.h>
#include <math.h>
#include <stdint.h>

typedef __attribute__((ext_vector_type(2))) float v2f;
typedef __attribute__((ext_vector_type(8))) float v8f;

#define NSPLIT 8
#define TOPK 8

// ---------------- top-k insertion network (register resident) ----------------
__device__ __forceinline__ void topk_insert(float (&tv)[TOPK], int (&ti)[TOPK],
                                            float v, int id) {
#pragma unroll
  for (int s = 0; s < TOPK; ++s) {
    bool better = (v > tv[s]) || ((v == tv[s]) && (id < ti[s]));
    float nv = better ? tv[s] : v;
    int   ni = better ? ti[s] : id;
    tv[s] = better ? v : tv[s];
    ti[s] = better ? id : ti[s];
    v = nv; id = ni;
  }
}

// ---------------- K1: per-gaussian setup ----------------
__global__ void k1_gauss_setup(const float* __restrict__ pos,
                               const float* __restrict__ scl,
                               const float* __restrict__ quat,
                               float* __restrict__ gfeat,
                               float* __restrict__ mnx, float* __restrict__ mny,
                               float* __restrict__ mnz, float* __restrict__ mxx,
                               float* __restrict__ mxy, float* __restrict__ mxz,
                               float* __restrict__ rad0, int G) {
  int g = blockIdx.x * blockDim.x + threadIdx.x;
  if (g >= G) return;
  float qw = quat[4 * g + 0], qx = quat[4 * g + 1];
  float qy = quat[4 * g + 2], qz = quat[4 * g + 3];
  float inv = rsqrtf(qw * qw + qx * qx + qy * qy + qz * qz);
  qw *= inv; qx *= inv; qy *= inv; qz *= inv;
  float r00 = 1.f - 2.f * (qy * qy + qz * qz), r01 = 2.f * (qx * qy - qw * qz),
        r02 = 2.f * (qx * qz + qw * qy);
  float r10 = 2.f * (qx * qy + qw * qz), r11 = 1.f - 2.f * (qx * qx + qz * qz),
        r12 = 2.f * (qy * qz - qw * qx);
  float r20 = 2.f * (qx * qz - qw * qy), r21 = 2.f * (qy * qz + qw * qx),
        r22 = 1.f - 2.f * (qx * qx + qy * qy);
  float s0 = scl[3 * g], s1 = scl[3 * g + 1], s2 = scl[3 * g + 2];
  float v0 = s0 * s0, v1 = s1 * s1, v2 = s2 * s2;
  // cov = R diag(v) R^T (symmetric)
  float a = r00 * r00 * v0 + r01 * r01 * v1 + r02 * r02 * v2;
  float b = r00 * r10 * v0 + r01 * r11 * v1 + r02 * r12 * v2;
  float c = r00 * r20 * v0 + r01 * r21 * v1 + r02 * r22 * v2;
  float d = r10 * r10 * v0 + r11 * r11 * v1 + r12 * r12 * v2;
  float e = r10 * r20 * v0 + r11 * r21 * v1 + r12 * r22 * v2;
  float f = r20 * r20 * v0 + r21 * r21 * v1 + r22 * r22 * v2;
  // adjugate inverse
  float C00 = d * f - e * e;
  float C01 = -(b * f - e * c);
  float C02 = b * e - d * c;
  float C11 = a * f - c * c;
  float C12 = -(a * e - b * c);
  float C22 = a * d - b * b;
  float det = a * C00 + b * C01 + c * C02;
  float idt = 1.0f / det;
  float A00 = C00 * idt, A01 = C01 * idt, A02 = C02 * idt;
  float A11 = C11 * idt, A12 = C12 * idt, A22 = C22 * idt;
  float px = pos[3 * g], py = pos[3 * g + 1], pz = pos[3 * g + 2];
  float Am0 = A00 * px + A01 * py + A02 * pz;
  float Am1 = A01 * px + A11 * py + A12 * pz;
  float Am2 = A02 * px + A12 * py + A22 * pz;
  float mAm = px * Am0 + py * Am1 + pz * Am2;
  float* gf = gfeat + (size_t)g * 16;
  gf[0] = A00; gf[1] = A01; gf[2] = A02;
  gf[3] = A01; gf[4] = A11; gf[5] = A12;
  gf[6] = A02; gf[7] = A12; gf[8] = A22;
  gf[9] = -2.f * Am0; gf[10] = -2.f * Am1; gf[11] = -2.f * Am2;
  gf[12] = mAm; gf[13] = 0.f; gf[14] = 0.f; gf[15] = 0.f;
  const float CS = 2.7954834f;
  float R0 = CS * sqrtf(a), R1 = CS * sqrtf(d), R2 = CS * sqrtf(f);
  mnx[g] = px - R0; mny[g] = py - R1; mnz[g] = pz - R2;
  mxx[g] = px + R0; mxy[g] = py + R1; mxz[g] = pz + R2;
  rad0[g] = R0;
}

// ---------------- K2: quantiles + median via LDS bitonic sort ----------------
__global__ void k2_stats(const float* __restrict__ mnx, const float* __restrict__ mny,
                         const float* __restrict__ mnz, const float* __restrict__ mxx,
                         const float* __restrict__ mxy, const float* __restrict__ mxz,
                         const float* __restrict__ rad0, float* __restrict__ scal,
                         int G) {
  extern __shared__ float s[];
  const float* arrs[7] = {mnx, mny, mnz, mxx, mxy, mxz, rad0};
  float res[7];
  for (int a = 0; a < 7; ++a) {
    for (int i = threadIdx.x; i < G; i += blockDim.x) s[i] = arrs[a][i];
    __syncthreads();
    for (int k = 2; k <= G; k <<= 1) {
      for (int j = k >> 1; j > 0; j >>= 1) {
        for (int i = threadIdx.x; i < G; i += blockDim.x) {
          int ixj = i ^ j;
          if (ixj > i) {
            float x = s[i], y = s[ixj];
            bool up = ((i & k) == 0);
            if ((x > y) == up) { s[i] = y; s[ixj] = x; }
          }
        }
        __syncthreads();
      }
    }
    if (threadIdx.x == 0) {
      float q = (a < 3) ? 0.01f : ((a < 6) ? 0.99f : 0.5f);
      float h = q * (float)(G - 1);
      int lo = (int)floorf(h);
      int hi = (lo + 1 < G) ? lo + 1 : G - 1;
      float fr = h - (float)lo;
      res[a] = s[lo] + fr * (s[hi] - s[lo]);
    }
    __syncthreads();
  }
  if (threadIdx.x == 0) {
    const float PADF = 0.1f;
    float gmin0 = res[0], gmin1 = res[1], gmin2 = res[2];
    float gmax0 = res[3], gmax1 = res[4], gmax2 = res[5];
    float sz0 = gmax0 - gmin0, sz1 = gmax1 - gmin1, sz2 = gmax2 - gmin2;
    scal[0] = gmin0 - PADF * sz0;
    scal[1] = gmin1 - PADF * sz1;
    scal[2] = gmin2 - PADF * sz2;
    scal[3] = gmax0 + PADF * sz0;
    scal[4] = gmax1 + PADF * sz1;
    scal[5] = gmax2 + PADF * sz2;
    scal[6] = res[6] * 3.0f;  // voxel = median(radius0) * VOXEL_SIZE_FACTOR
  }
}

// ---------------- K3: per-gaussian packed voxel box (16B) ----------------
// Oversized gaussians get a box widened to [-32768,32767]^3 so eligibility
// reduces to a pure range test (no flag).
__global__ void k3_gbox(const float* __restrict__ mnx, const float* __restrict__ mny,
                        const float* __restrict__ mnz, const float* __restrict__ mxx,
                        const float* __restrict__ mxy, const float* __restrict__ mxz,
                        const float* __restrict__ scal, int4* __restrict__ gbox, int G) {
  int g = blockIdx.x * blockDim.x + threadIdx.x;
  if (g >= G) return;
  float iv = 1.0f / scal[6];
  int v0 = (int)floorf(mnx[g] * iv);
  int v1 = (int)floorf(mny[g] * iv);
  int v2 = (int)floorf(mnz[g] * iv);
  int w0 = (int)floorf(mxx[g] * iv);
  int w1 = (int)floorf(mxy[g] * iv);
  int w2 = (int)floorf(mxz[g] * iv);
  int nvox = (w0 - v0 + 1) * (w1 - v1 + 1) * (w2 - v2 + 1);
  if (nvox > 64) {  // OVERSIZED_THRESH: always eligible
    v0 = v1 = v2 = -32768;
    w0 = w1 = w2 = 32767;
  }
  int x0 = (v0 & 0xffff) | (v1 << 16);
  int x1 = (v2 & 0xffff) | (w0 << 16);
  int x2 = (w1 & 0xffff) | (w2 << 16);
  gbox[g] = make_int4(x0, x1, x2, 0);
}

// ---------------- K4: morton argsort (stable via key<<32|idx packing) ----------------
__device__ __forceinline__ unsigned part1by2(unsigned x) {
  x &= 1023u;
  x = (x | (x << 16)) & 50331903u;
  x = (x | (x << 8)) & 50393103u;
  x = (x | (x << 4)) & 51130563u;
  x = (x | (x << 2)) & 153391689u;
  return x;
}

__global__ void k4_morton_sort(const float* __restrict__ pc,
                               const float* __restrict__ scal,
                               int* __restrict__ order, int P) {
  extern __shared__ unsigned long long su[];
  float g0 = scal[0], g1 = scal[1], g2 = scal[2];
  float i0 = 1.0f / (scal[3] - g0);
  float i1 = 1.0f / (scal[4] - g1);
  float i2 = 1.0f / (scal[5] - g2);
  const float CL = 0.99999994f;  // 1 - 1e-7 (f32)
  for (int i = threadIdx.x; i < P; i += blockDim.x) {
    float x = pc[3 * i], y = pc[3 * i + 1], z = pc[3 * i + 2];
    float ux = fminf(fmaxf((x - g0) * i0, 0.0f), CL);
    float uy = fminf(fmaxf((y - g1) * i1, 0.0f), CL);
    float uz = fminf(fmaxf((z - g2) * i2, 0.0f), CL);
    unsigned cx = (unsigned)(ux * 1024.0f);
    unsigned cy = (unsigned)(uy * 1024.0f);
    unsigned cz = (unsigned)(uz * 1024.0f);
    unsigned code = part1by2(cx) | (part1by2(cy) << 1) | (part1by2(cz) << 2);
    su[i] = ((unsigned long long)code << 32) | (unsigned)i;
  }
  __syncthreads();
  for (int k = 2; k <= P; k <<= 1) {
    for (int j = k >> 1; j > 0; j >>= 1) {
      for (int i = threadIdx.x; i < P; i += blockDim.x) {
        int ixj = i ^ j;
        if (ixj > i) {
          unsigned long long A = su[i], B = su[ixj];
          bool up = ((i & k) == 0);
          if ((A > B) == up) { su[i] = B; su[ixj] = A; }
        }
      }
      __syncthreads();
    }
  }
  for (int i = threadIdx.x; i < P; i += blockDim.x)
    order[i] = (int)(su[i] & 0xffffffffu);
}

// ---------------- K5: gather sorted points, build point features ----------------
__global__ void k5_pfeat(const float* __restrict__ pc, const int* __restrict__ order,
                         const float* __restrict__ scal, float* __restrict__ out_sp,
                         float* __restrict__ pfeat, int4* __restrict__ pvox, int P) {
  int p = blockIdx.x * blockDim.x + threadIdx.x;
  if (p >= P) return;
  int o = order[p];
  float x = pc[3 * o], y = pc[3 * o + 1], z = pc[3 * o + 2];
  out_sp[3 * p] = x; out_sp[3 * p + 1] = y; out_sp[3 * p + 2] = z;
  float* pf = pfeat + (size_t)p * 16;
  pf[0] = x * x; pf[1] = x * y; pf[2] = x * z;
  pf[3] = y * x; pf[4] = y * y; pf[5] = y * z;
  pf[6] = z * x; pf[7] = z * y; pf[8] = z * z;
  pf[9] = x; pf[10] = y; pf[11] = z; pf[12] = 1.0f;
  pf[13] = 0.f; pf[14] = 0.f; pf[15] = 0.f;
  float iv = 1.0f / scal[6];
  pvox[p] = make_int4((int)floorf(x * iv), (int)floorf(y * iv),
                      (int)floorf(z * iv), 0);
}

// ---------------- K6: WMMA score matrix + per-lane top-8 ----------------
// Block = 8 waves, all sharing one gaussian split. Gaussian tiles (features +
// packed boxes) are double-buffer prefetched into LDS with
// global_load_async_to_lds_b128 (ASYNCcnt) and consumed via ds loads, so the
// 4 chained V_WMMA_F32_16X16X4_F32 read short-latency LDS instead of global.
// LDS feature rows are padded to 18 dwords -> conflict-free, 8B-aligned b64s.
__global__ void __launch_bounds__(256)
k6_score_topk(const float* __restrict__ gfeat, const int4* __restrict__ gbox,
              const float* __restrict__ pfeat, const int4* __restrict__ pvox,
              float* __restrict__ part_vals, int* __restrict__ part_ids,
              int G, int P) {
  __shared__ float sgf[2][16 * 18];
  __shared__ int4 sbox[2][16];

  int tid = threadIdx.x;
  int lane = tid & 31;
  int waveid = tid >> 5;
  int split = blockIdx.x % NSPLIT;
  int bg = blockIdx.x / NSPLIT;
  int ptile = bg * 8 + waveid;  // requires P % 128 == 0
  int n = lane & 15;
  int h = lane >> 4;
  int p = ptile * 16 + n;

  int gChunk = G / NSPLIT;
  int gBegin = split * gChunk;
  int nT = gChunk / 16;

  // async-stage one 16-gaussian tile (1KB feat + 256B boxes) into buffer `buf`
  auto stage = [&](int buf, int g0) {
    if (tid < 64) {
      int r = tid >> 2, c = tid & 3;
      const float* src = gfeat + (size_t)(g0 + r) * 16 + c * 4;
      unsigned dst = (unsigned)(uintptr_t)&sgf[buf][r * 18 + c * 4];
      unsigned long long ga = (unsigned long long)(uintptr_t)src;
      asm volatile("global_load_async_to_lds_b128 %0, %1, off"
                   :: "v"(dst), "v"(ga) : "memory");
    } else if (tid < 80) {
      int i = tid - 64;
      const int4* src = gbox + (g0 + i);
      unsigned dst = (unsigned)(uintptr_t)&sbox[buf][i];
      unsigned long long ga = (unsigned long long)(uintptr_t)src;
      asm volatile("global_load_async_to_lds_b128 %0, %1, off"
                   :: "v"(dst), "v"(ga) : "memory");
    }
  };

  // B fragments (point features), fixed for the whole wave lifetime.
  // f32 16x16x4 operand layout: half-wave h holds K=2h+{0,1} in {v0,v1}.
  float2 bfrag[4];
  const float* pf = pfeat + (size_t)p * 16;
#pragma unroll
  for (int w = 0; w < 4; ++w)
    bfrag[w] = *(const float2*)(pf + 4 * w + 2 * h);
  int4 pv = pvox[p];

  float tv[TOPK]; int ti[TOPK];
#pragma unroll
  for (int i = 0; i < TOPK; ++i) { tv[i] = -1.0f; ti[i] = 0x7fffffff; }

  stage(0, gBegin);
  asm volatile("s_wait_asynccnt 0x0" ::: "memory");
  __syncthreads();

  for (int t = 0; t < nT; ++t) {
    int g0 = gBegin + t * 16;
    if (t + 1 < nT) stage((t + 1) & 1, g0 + 16);

    const float* sg = &sgf[t & 1][0];
    float2 afr[4];
#pragma unroll
    for (int w = 0; w < 4; ++w)
      afr[w] = *(const float2*)(sg + (lane & 15) * 18 + 4 * w + 2 * h);
    v8f cacc = {};
#pragma unroll
    for (int w = 0; w < 4; ++w) {
      v2f a = {afr[w].x, afr[w].y};
      v2f b = {bfrag[w].x, bfrag[w].y};
      cacc = __builtin_amdgcn_wmma_f32_16x16x4_f32(
          false, a, false, b, (short)0, cacc, false, false);
    }
    // cacc[r] = maha(point p, gaussian g0 + 8h + r)
#pragma unroll
    for (int r = 0; r < 8; ++r) {
      int g = g0 + 8 * h + r;
      int4 B = sbox[t & 1][8 * h + r];  // broadcast read per half-wave
      int mnxv = (int)(short)(B.x & 0xffff), mnyv = B.x >> 16;
      int mnzv = (int)(short)(B.y & 0xffff), mxxv = B.y >> 16;
      int mxyv = (int)(short)(B.z & 0xffff), mxzv = B.z >> 16;
      bool elig = (pv.x >= mnxv) & (pv.x <= mxxv) & (pv.y >= mnyv) &
                  (pv.y <= mxyv) & (pv.z >= mnzv) & (pv.z <= mxzv);
      float sc = elig ? __expf(-0.5f * cacc[r]) : 0.0f;
      // streaming ids are strictly ascending per lane -> strict > is enough
      if (sc > tv[TOPK - 1]) topk_insert(tv, ti, sc, g);
    }

    asm volatile("s_wait_asynccnt 0x0" ::: "memory");
    __syncthreads();
  }

  // merge lane pair (l <-> l^16): both see same point, disjoint gaussian rows
#pragma unroll
  for (int i = 0; i < TOPK; ++i) {
    float ov = __shfl_xor(tv[i], 16, 32);
    int oi = __shfl_xor(ti[i], 16, 32);
    if ((ov > tv[TOPK - 1]) ||
        ((ov == tv[TOPK - 1]) && (oi < ti[TOPK - 1])))
      topk_insert(tv, ti, ov, oi);
  }

  if (h == 0) {
    float* pvd = part_vals + ((size_t)p * NSPLIT + split) * TOPK;
    int* pid = part_ids + ((size_t)p * NSPLIT + split) * TOPK;
#pragma unroll
    for (int i = 0; i < TOPK; ++i) { pvd[i] = tv[i]; pid[i] = ti[i]; }
  }
}

// ---------------- K7: merge splits, write ids_f and vals ----------------
__global__ void k7_merge(const float* __restrict__ part_vals,
                         const int* __restrict__ part_ids,
                         float* __restrict__ out, int P) {
  int p = blockIdx.x * blockDim.x + threadIdx.x;
  if (p >= P) return;
  float tv[TOPK]; int ti[TOPK];
#pragma unroll
  for (int i = 0; i < TOPK; ++i) { tv[i] = -1.0f; ti[i] = 0x7fffffff; }
  for (int s = 0; s < NSPLIT; ++s) {
    const float* pvd = part_vals + ((size_t)p * NSPLIT + s) * TOPK;
    const int* pid = part_ids + ((size_t)p * NSPLIT + s) * TOPK;
#pragma unroll
    for (int i = 0; i < TOPK; ++i) {
      float v = pvd[i]; int id = pid[i];
      if ((v > tv[TOPK - 1]) || ((v == tv[TOPK - 1]) && (id < ti[TOPK - 1])))
        topk_insert(tv, ti, v, id);
    }
  }
  float* idsOut = out + (size_t)3 * P;
  float* valsOut = out + (size_t)11 * P;
#pragma unroll
  for (int i = 0; i < TOPK; ++i) {
    idsOut[(size_t)p * TOPK + i] = (tv[i] > 0.0f) ? (float)ti[i] : -1.0f;
    valsOut[(size_t)p * TOPK + i] = tv[i];
  }
}

// ---------------- launch ----------------
extern "C" void kernel_launch(void* const* d_in, const int* in_sizes, int n_in,
                              void* d_out, int out_size, void* d_ws, size_t ws_size,
                              hipStream_t stream) {
  const float* positions = (const float*)d_in[0];
  const float* scales = (const float*)d_in[1];
  const float* quats = (const float*)d_in[2];
  const float* pc = (const float*)d_in[3];
  int G = in_sizes[0] / 3;
  int P = in_sizes[3] / 3;
  float* out = (float*)d_out;

  uintptr_t base = (uintptr_t)d_ws;
  auto alloc = [&](size_t bytes) -> void* {
    void* ptr = (void*)base;
    base += (bytes + 255) & ~(size_t)255;
    return ptr;
  };
  float* gfeat = (float*)alloc((size_t)G * 16 * 4);
  int4* gbox = (int4*)alloc((size_t)G * 16);
  float* mnx = (float*)alloc((size_t)G * 4);
  float* mny = (float*)alloc((size_t)G * 4);
  float* mnz = (float*)alloc((size_t)G * 4);
  float* mxx = (float*)alloc((size_t)G * 4);
  float* mxy = (float*)alloc((size_t)G * 4);
  float* mxz = (float*)alloc((size_t)G * 4);
  float* rad0 = (float*)alloc((size_t)G * 4);
  float* scal = (float*)alloc(64);
  int* order = (int*)alloc((size_t)P * 4);
  float* pfeat = (float*)alloc((size_t)P * 16 * 4);
  int4* pvox = (int4*)alloc((size_t)P * 4 * 4);
  float* part_vals = (float*)alloc((size_t)P * NSPLIT * TOPK * 4);
  int* part_ids = (int*)alloc((size_t)P * NSPLIT * TOPK * 4);

  k1_gauss_setup<<<(G + 255) / 256, 256, 0, stream>>>(
      positions, scales, quats, gfeat, mnx, mny, mnz, mxx, mxy, mxz, rad0, G);
  k2_stats<<<1, 1024, (size_t)G * 4, stream>>>(mnx, mny, mnz, mxx, mxy, mxz,
                                               rad0, scal, G);
  k3_gbox<<<(G + 255) / 256, 256, 0, stream>>>(mnx, mny, mnz, mxx, mxy, mxz,
                                               scal, gbox, G);
  k4_morton_sort<<<1, 1024, (size_t)P * 8, stream>>>(pc, scal, order, P);
  k5_pfeat<<<(P + 255) / 256, 256, 0, stream>>>(pc, order, scal, out, pfeat,
                                                pvox, P);
  int nBlocks = (P / 128) * NSPLIT;  // 8 waves/block, each wave = 16 points
  k6_score_topk<<<nBlocks, 256, 0, stream>>>(gfeat, gbox, pfeat, pvox,
                                             part_vals, part_ids, G, P);
  k7_merge<<<(P + 255) / 256, 256, 0, stream>>>(part_vals, part_ids, out, P);
}